// SparseUnifiedOutputLoss_15479062134913
// MI455X (gfx1250) — compile-verified
//
#include <hip/hip_runtime.h>

typedef __attribute__((ext_vector_type(4))) float f4;
typedef __attribute__((ext_vector_type(2))) float v2f;
typedef __attribute__((ext_vector_type(8))) float v8f;

// ---------------------------------------------------------------------------
// Wave32 sum via V_WMMA_F32_16X16X4_F32 with an all-ones B matrix.
// A holds one partial per lane (A.y = 0); with B == ones,
// D[m][n] = sum_k A[m][k], replicated across all 16 N columns.
// Per documented C/D layout, sum of a lane's 8 D VGPRs = sum over M rows 0-7
// (lanes 0-15) or 8-15 (lanes 16-31); one xor-16 shuffle gives the wave total.
// Requires EXEC == all ones (callers guarantee: full blocks, no divergence).
// ---------------------------------------------------------------------------
__device__ __forceinline__ float wave_sum_wmma(float x) {
  v2f A; A[0] = x;    A[1] = 0.0f;
  v2f B; B[0] = 1.0f; B[1] = 1.0f;
  v8f C = {0.f, 0.f, 0.f, 0.f, 0.f, 0.f, 0.f, 0.f};
  C = __builtin_amdgcn_wmma_f32_16x16x4_f32(false, A, false, B,
                                            (short)0, C, false, false);
  float s = ((C[0] + C[1]) + (C[2] + C[3])) + ((C[4] + C[5]) + (C[6] + C[7]));
  s += __shfl_xor(s, 16, 32);
  return s;  // full wave sum, identical in every lane
}

// Block reduction: wmma wave sums -> LDS -> wave 0 shuffle tree.
// Result valid for wave 0 (thread 0 uses it). blockDim.x must be 256.
__device__ __forceinline__ float block_sum(float x, float* lds) {
  float s = wave_sum_wmma(x);
  const int lane = threadIdx.x & 31;
  const int wv   = threadIdx.x >> 5;
  if (lane == 0) lds[wv] = s;
  __syncthreads();
  float v = 0.0f;
  if (wv == 0) {
    v = (lane < (int)(blockDim.x >> 5)) ? lds[lane] : 0.0f;
    v += __shfl_xor(v, 1, 32);
    v += __shfl_xor(v, 2, 32);
    v += __shfl_xor(v, 4, 32);
  }
  return v;
}

// Per-element loss term, MSE_LEAK = 0, LOGIT_LEAK = 0.5:
//   ola = m * (0.5*ol + lk),  A = sq*m + o*o - 2*gt*m*o,  val = A*ola*w
__device__ __forceinline__ float elem4(f4 g, f4 q, f4 w, f4 o, f4 m, f4 l,
                                       float lk) {
  float r = 0.0f;
#pragma unroll
  for (int c = 0; c < 4; ++c) {
    float ola = m[c] * fmaf(l[c], 0.5f, lk);
    float A   = fmaf(o[c], o[c], q[c] * m[c]) - 2.0f * g[c] * (m[c] * o[c]);
    r = fmaf(A * ola, w[c], r);
  }
  return r;
}

struct Group {
  const float* o;
  const float* m;
  const float* ol;
  float coef;  // local_mult * (1/ALPHA)^d / total_mult
  float lk;    // 0.5 / num_nn
};

struct LevelArgs {
  const float* gt;
  const float* sq;
  const float* w;
  Group g0, g1, g2;
  int ngroups;      // 2 or 3
  int nvec;         // element count / 4
  float* partials;  // one float per block
};

__global__ __launch_bounds__(256) void level_loss_kernel(LevelArgs a) {
  __shared__ float lds[8];
  const int stride = gridDim.x * blockDim.x;
  const f4* gt = (const f4*)a.gt;
  const f4* sq = (const f4*)a.sq;
  const f4* w  = (const f4*)a.w;
  const f4* o0 = (const f4*)a.g0.o; const f4* m0 = (const f4*)a.g0.m; const f4* l0 = (const f4*)a.g0.ol;
  const f4* o1 = (const f4*)a.g1.o; const f4* m1 = (const f4*)a.g1.m; const f4* l1 = (const f4*)a.g1.ol;
  const f4* o2 = (const f4*)a.g2.o; const f4* m2 = (const f4*)a.g2.m; const f4* l2 = (const f4*)a.g2.ol;
  const bool has2 = (a.ngroups > 2);  // uniform across the block

  float acc0 = 0.0f, acc1 = 0.0f, acc2 = 0.0f;
  for (int i = blockIdx.x * blockDim.x + threadIdx.x; i < a.nvec; i += stride) {
    // Streaming one-pass data: non-temporal b128 loads (keep L2 clean).
    f4 g = __builtin_nontemporal_load(&gt[i]);
    f4 q = __builtin_nontemporal_load(&sq[i]);
    f4 ww = __builtin_nontemporal_load(&w[i]);
    {
      f4 o = __builtin_nontemporal_load(&o0[i]);
      f4 m = __builtin_nontemporal_load(&m0[i]);
      f4 l = __builtin_nontemporal_load(&l0[i]);
      acc0 += elem4(g, q, ww, o, m, l, a.g0.lk);
    }
    {
      f4 o = __builtin_nontemporal_load(&o1[i]);
      f4 m = __builtin_nontemporal_load(&m1[i]);
      f4 l = __builtin_nontemporal_load(&l1[i]);
      acc1 += elem4(g, q, ww, o, m, l, a.g1.lk);
    }
    if (has2) {
      f4 o = __builtin_nontemporal_load(&o2[i]);
      f4 m = __builtin_nontemporal_load(&m2[i]);
      f4 l = __builtin_nontemporal_load(&l2[i]);
      acc2 += elem4(g, q, ww, o, m, l, a.g2.lk);
    }
  }
  float acc = fmaf(a.g0.coef, acc0,
                   fmaf(a.g1.coef, acc1, a.g2.coef * acc2));
  float v = block_sum(acc, lds);
  if (threadIdx.x == 0) a.partials[blockIdx.x] = v;
}

__global__ __launch_bounds__(256) void finalize_kernel(const float* partials,
                                                       int n, float* out) {
  __shared__ float lds[8];
  float acc = 0.0f;
  for (int i = threadIdx.x; i < n; i += (int)blockDim.x) acc += partials[i];
  float v = block_sum(acc, lds);
  if (threadIdx.x == 0) out[0] = v;
}

static inline int blocks_for(int nvec, int cap) {
  int b = (nvec + 255) / 256;
  return b < cap ? b : cap;
}

extern "C" void kernel_launch(void* const* d_in, const int* in_sizes, int n_in,
                              void* d_out, int out_size, void* d_ws,
                              size_t ws_size, hipStream_t stream) {
  (void)n_in; (void)out_size; (void)ws_size;
  // setup_inputs() dict order:
  //  0:gt0 1:sq0 2:w0  3:gt1 4:sq1 5:w1  6:gt2 7:sq2 8:w2
  //  9:po1 10:pm1 11:pl1  12:po2 13:pm2 14:pl2
  // 15:to0 16:tm0 17:tl0  18:to1 19:tm1 20:tl1  21:to2 22:tm2 23:tl2
  // 24:no0 25:nm0 26:nl0  27:no1 28:nm1 29:nl1
  const float* gt0 = (const float*)d_in[0];
  const float* sq0 = (const float*)d_in[1];
  const float* w0  = (const float*)d_in[2];
  const float* gt1 = (const float*)d_in[3];
  const float* sq1 = (const float*)d_in[4];
  const float* w1  = (const float*)d_in[5];
  const float* gt2 = (const float*)d_in[6];
  const float* sq2 = (const float*)d_in[7];
  const float* w2  = (const float*)d_in[8];

  const int N0 = in_sizes[0], N1 = in_sizes[3], N2 = in_sizes[6];
  const int nv0 = N0 >> 2, nv1 = N1 >> 2, nv2 = N2 >> 2;
  const int b0 = blocks_for(nv0, 2048);
  const int b1 = blocks_for(nv1, 1024);
  const int b2 = blocks_for(nv2, 256);

  // coef = local_mult[dd] * (1/ALPHA)^d / total_mult; total_mult = 16+4+1 = 21
  const double INV = 1.0 / 0.9, NORM = 1.0 / 21.0;
  const float c_this0 = (float)(1.0 * NORM);              // d0,dd1
  const float c_prev1 = (float)(0.9 * INV * NORM);        // d1,dd0
  const float c_nxt0  = (float)(INV * NORM);              // d0,dd2
  const float c_this1 = (float)(INV * NORM);              // d1,dd1
  const float c_prev2 = (float)(0.9 * INV * INV * NORM);  // d2,dd0
  const float c_nxt1  = (float)(INV * INV * NORM);        // d1,dd2
  const float c_this2 = (float)(INV * INV * NORM);        // d2,dd1
  const float lk2 = 0.25f;                 // 0.5 / num_nn, num_nn=2 (d=0, d=2)
  const float lk3 = (float)(0.5 / 3.0);    // num_nn=3 (d=1)

  float* ws = (float*)d_ws;

  // Level 0 (1024x1024): this[0] (d0,dd1, lk 0.25) + prev[1] (d1,dd0, lk 1/6)
  LevelArgs a0;
  a0.gt = gt0; a0.sq = sq0; a0.w = w0;
  a0.g0 = Group{(const float*)d_in[15], (const float*)d_in[16],
                (const float*)d_in[17], c_this0, lk2};
  a0.g1 = Group{(const float*)d_in[9], (const float*)d_in[10],
                (const float*)d_in[11], c_prev1, lk3};
  a0.g2 = Group{nullptr, nullptr, nullptr, 0.0f, 0.0f};
  a0.ngroups = 2; a0.nvec = nv0; a0.partials = ws;

  // Level 1 (512x512): nxt[0] (d0,dd2, lk 0.25) + this[1] (d1,dd1, lk 1/6)
  //                    + prev[2] (d2,dd0, lk 0.25)
  LevelArgs a1;
  a1.gt = gt1; a1.sq = sq1; a1.w = w1;
  a1.g0 = Group{(const float*)d_in[24], (const float*)d_in[25],
                (const float*)d_in[26], c_nxt0, lk2};
  a1.g1 = Group{(const float*)d_in[18], (const float*)d_in[19],
                (const float*)d_in[20], c_this1, lk3};
  a1.g2 = Group{(const float*)d_in[12], (const float*)d_in[13],
                (const float*)d_in[14], c_prev2, lk2};
  a1.ngroups = 3; a1.nvec = nv1; a1.partials = ws + b0;

  // Level 2 (256x256): nxt[1] (d1,dd2, lk 1/6) + this[2] (d2,dd1, lk 0.25)
  LevelArgs a2;
  a2.gt = gt2; a2.sq = sq2; a2.w = w2;
  a2.g0 = Group{(const float*)d_in[27], (const float*)d_in[28],
                (const float*)d_in[29], c_nxt1, lk3};
  a2.g1 = Group{(const float*)d_in[21], (const float*)d_in[22],
                (const float*)d_in[23], c_this2, lk2};
  a2.g2 = Group{nullptr, nullptr, nullptr, 0.0f, 0.0f};
  a2.ngroups = 2; a2.nvec = nv2; a2.partials = ws + b0 + b1;

  level_loss_kernel<<<dim3(b0), dim3(256), 0, stream>>>(a0);
  level_loss_kernel<<<dim3(b1), dim3(256), 0, stream>>>(a1);
  level_loss_kernel<<<dim3(b2), dim3(256), 0, stream>>>(a2);
  finalize_kernel<<<dim3(1), dim3(256), 0, stream>>>(ws, b0 + b1 + b2,
                                                     (float*)d_out);
}